// M12_70480413327935
// MI455X (gfx1250) — compile-verified
//
#include <hip/hip_runtime.h>
#include <hip/hip_bf16.h>

typedef __attribute__((ext_vector_type(2))) float v2f;
typedef __attribute__((ext_vector_type(8))) float v8f;

#define D64 64
#define ED16 16
#define M128 128

static __device__ __forceinline__ float leaky(float x) {
    return x > 0.0f ? x : 0.01f * x;
}

static __device__ __forceinline__ v8f wmma4(v2f a, v2f b, v8f c) {
    // D = A(16x4,f32) * B(4x16,f32) + C(16x16,f32)
    return __builtin_amdgcn_wmma_f32_16x16x4_f32(
        false, a, false, b, (short)0, c, false, false);
}

// ---------------------------------------------------------------- utilities
__global__ void fill_zero_kernel(float* __restrict__ p, int n) {
    int i = blockIdx.x * blockDim.x + threadIdx.x;
    if (i < n) p[i] = 0.0f;
}

// ------------------------------------------------------------- BN statistics
// Accumulate per-feature sum and sum-of-squares over N rows (C features).
template <int C>
__global__ void bn_stats_kernel(const float* __restrict__ x, int n,
                                float* __restrict__ gsum,
                                float* __restrict__ gsumsq) {
    const int RPB = 256 / C;               // rows processed per block per iter
    int f = threadIdx.x % C;
    int rsub = threadIdx.x / C;
    float s = 0.0f, s2 = 0.0f;
    for (int r = blockIdx.x * RPB + rsub; r < n; r += gridDim.x * RPB) {
        float v = x[(size_t)r * C + f];
        s += v;
        s2 += v * v;
    }
    __shared__ float ls[256];
    __shared__ float ls2[256];
    ls[threadIdx.x] = s;
    ls2[threadIdx.x] = s2;
    __syncthreads();
    if (rsub == 0) {
        for (int j = 1; j < RPB; j++) {
            s += ls[j * C + f];
            s2 += ls2[j * C + f];
        }
        unsafeAtomicAdd(&gsum[f], s);
        unsafeAtomicAdd(&gsumsq[f], s2);
    }
}

// scale = w * rsqrt(var + eps); shift = b - mean*scale   (biased batch stats)
__global__ void bn_finalize_kernel(const float* __restrict__ gsum,
                                   const float* __restrict__ gsumsq, int n,
                                   const float* __restrict__ w,
                                   const float* __restrict__ b,
                                   float* __restrict__ scale,
                                   float* __restrict__ shift) {
    int f = threadIdx.x;
    float inv_n = 1.0f / (float)n;
    float mean = gsum[f] * inv_n;
    float var = gsumsq[f] * inv_n - mean * mean;
    float sc = w[f] * rsqrtf(var + 1e-5f);
    scale[f] = sc;
    shift[f] = b[f] - mean * sc;
}

// ----------------------------------------------------------------- node prep
// hi = act(bn(h)); aggr = (1+eps)*hi   (edge kernel then atomically adds msgs)
__global__ void node_prep_kernel(const float* __restrict__ hin,
                                 const float* __restrict__ scale,
                                 const float* __restrict__ shift,
                                 const float* __restrict__ epsArr, int layer,
                                 float* __restrict__ hi,
                                 float* __restrict__ aggr, int total) {
    int idx = blockIdx.x * blockDim.x + threadIdx.x;
    if (idx >= total) return;
    float ep1 = 1.0f + epsArr[layer];
    int f = idx & 63;
    float v = hin[idx] * scale[f] + shift[f];
    if (layer > 0) v = leaky(v);
    hi[idx] = v;
    aggr[idx] = ep1 * v;
}

// ---------------------------------------------------------------- edge kernel
// Per wave: 16 edges. e = edge_attr(16x16) @ We^T(16x64) via 16 WMMAs,
// msg = relu(hi[src] + e + bias), atomic scatter-add into aggr[dst].
__global__ void edge_kernel(const float* __restrict__ eattr,
                            const int* __restrict__ src,
                            const int* __restrict__ dst,
                            const float* __restrict__ We,   // [64][16]
                            const float* __restrict__ be,   // [64]
                            const float* __restrict__ hi,
                            float* __restrict__ aggr, int nTiles) {
    __shared__ float2 sB[8 * 64];   // [kpair(8)][n(64)] of (B[k][n],B[k+1][n])
    __shared__ float sBias[64];
    int tid = threadIdx.x;
    for (int idx = tid; idx < 8 * 64; idx += 256) {
        int kp = idx >> 6;
        int n = idx & 63;
        // B[k][n] = We[n][k]
        sB[idx] = make_float2(We[n * ED16 + kp * 2], We[n * ED16 + kp * 2 + 1]);
    }
    if (tid < 64) sBias[tid] = be[tid];
    __syncthreads();

    int lane = tid & 31;
    int wave = tid >> 5;
    int half = lane >> 4;   // 0: lanes 0-15, 1: lanes 16-31
    int l15 = lane & 15;

    // Preload B fragments (uniform across tiles): kk in 0..3, ntile in 0..3
    v2f bf[4][4];
#pragma unroll
    for (int kk = 0; kk < 4; kk++)
#pragma unroll
        for (int t = 0; t < 4; t++)
            bf[kk][t] = *(const v2f*)&sB[(kk * 2 + half) * 64 + t * 16 + l15];

    int gwave = blockIdx.x * 8 + wave;
    int nWaves = gridDim.x * 8;
    for (int tile = gwave; tile < nTiles; tile += nWaves) {
        int ebase = tile * 16;
        // lanes 0-15 hold src, lanes 16-31 hold dst of the 16 edges
        int idxReg = (half == 0) ? src[ebase + l15] : dst[ebase + l15];

        const float* ar = eattr + (size_t)(ebase + l15) * ED16 + half * 2;
        v2f a[4];
#pragma unroll
        for (int kk = 0; kk < 4; kk++) a[kk] = *(const v2f*)(ar + kk * 4);

        v8f acc[4];
#pragma unroll
        for (int t = 0; t < 4; t++) acc[t] = (v8f){0, 0, 0, 0, 0, 0, 0, 0};
#pragma unroll
        for (int kk = 0; kk < 4; kk++)
#pragma unroll
            for (int t = 0; t < 4; t++) acc[t] = wmma4(a[kk], bf[kk][t], acc[t]);

#pragma unroll
        for (int r = 0; r < 8; r++) {
            int row = half ? (r + 8) : r;        // this lane's matrix row
            int sidx = __shfl(idxReg, row);      // src node of edge `row`
            int didx = __shfl(idxReg, row + 16); // dst node of edge `row`
            const float* hrow = hi + (size_t)sidx * D64;
            float* arow = aggr + (size_t)didx * D64;
#pragma unroll
            for (int t = 0; t < 4; t++) {
                int col = t * 16 + l15;
                float v = acc[t][r] + sBias[col] + hrow[col];
                v = fmaxf(v, 0.0f);
                unsafeAtomicAdd(&arow[col], v);
            }
        }
    }
}

// ------------------------------------------------------------------ GEMM
// out[N x OUTC] = act(A) @ W^T + bias (+ resid). A is [N x 64].
// xform: A-element transform a -> leaky(a*scale[k] + shift[k]) (mid-BN fold).
template <int OUTC>
__global__ void gemm_wmma_kernel(const float* __restrict__ A,
                                 const float* __restrict__ W,     // [OUTC][64]
                                 const float* __restrict__ bias,  // [OUTC]
                                 const float* __restrict__ scale, // [64] or null
                                 const float* __restrict__ shift, // [64] or null
                                 const float* __restrict__ resid, // [N][OUTC] or null
                                 float* __restrict__ out, int nRowTiles,
                                 int xform) {
    constexpr int NT = OUTC / 16;
    __shared__ float2 sB[32 * OUTC];   // [kpair(32)][n]
    __shared__ float sBias[OUTC];
    __shared__ float sScale[64];
    __shared__ float sShift[64];
    int tid = threadIdx.x;
    for (int idx = tid; idx < 32 * OUTC; idx += 256) {
        int kp = idx / OUTC;
        int n = idx % OUTC;
        sB[idx] = make_float2(W[n * D64 + kp * 2], W[n * D64 + kp * 2 + 1]);
    }
    for (int idx = tid; idx < OUTC; idx += 256) sBias[idx] = bias[idx];
    if (xform && tid < 64) {
        sScale[tid] = scale[tid];
        sShift[tid] = shift[tid];
    }
    __syncthreads();

    int lane = tid & 31;
    int wave = tid >> 5;
    int half = lane >> 4;
    int l15 = lane & 15;
    int gwave = blockIdx.x * 8 + wave;
    int nWaves = gridDim.x * 8;

    for (int rt = gwave; rt < nRowTiles; rt += nWaves) {
        int rbase = rt * 16;
        const float* ar = A + (size_t)(rbase + l15) * D64 + half * 2;
        v8f acc[NT];
#pragma unroll
        for (int t = 0; t < NT; t++) acc[t] = (v8f){0, 0, 0, 0, 0, 0, 0, 0};

#pragma unroll
        for (int kk = 0; kk < 16; kk++) {
            v2f a = *(const v2f*)(ar + kk * 4);
            if (xform) {
                int k0 = kk * 4 + half * 2;
                a.x = leaky(a.x * sScale[k0] + sShift[k0]);
                a.y = leaky(a.y * sScale[k0 + 1] + sShift[k0 + 1]);
            }
#pragma unroll
            for (int t = 0; t < NT; t++) {
                v2f b = *(const v2f*)&sB[(kk * 2 + half) * OUTC + t * 16 + l15];
                acc[t] = wmma4(a, b, acc[t]);
            }
        }

#pragma unroll
        for (int r = 0; r < 8; r++) {
            int row = rbase + (half ? (r + 8) : r);
            float* orow = out + (size_t)row * OUTC;
            const float* rrow = resid ? resid + (size_t)row * OUTC : nullptr;
#pragma unroll
            for (int t = 0; t < NT; t++) {
                int col = t * 16 + l15;
                float v = acc[t][r] + sBias[col];
                if (rrow) v += rrow[col];
                orow[col] = v;
            }
        }
    }
}

// ----------------------------------------------------- classifier final dot
// y[n] = sum_m leaky(t[n][m]*scale[m]+shift[m]) * w2[m] + b2
__global__ void cls_final_kernel(const float* __restrict__ t,
                                 const float* __restrict__ scale,
                                 const float* __restrict__ shift,
                                 const float* __restrict__ w2,
                                 const float* __restrict__ b2,
                                 float* __restrict__ y, int n) {
    int lane = threadIdx.x & 31;
    int gwave = (blockIdx.x * blockDim.x + threadIdx.x) >> 5;
    int nWaves = (gridDim.x * blockDim.x) >> 5;
    for (int row = gwave; row < n; row += nWaves) {
        const float* tr = t + (size_t)row * M128;
        float acc = 0.0f;
#pragma unroll
        for (int j = 0; j < 4; j++) {
            int m = lane + j * 32;
            float v = leaky(tr[m] * scale[m] + shift[m]);
            acc += v * w2[m];
        }
#pragma unroll
        for (int o = 16; o > 0; o >>= 1) acc += __shfl_xor(acc, o);
        if (lane == 0) y[row] = acc + b2[0];
    }
}

// ------------------------------------------------------------------- driver
extern "C" void kernel_launch(void* const* d_in, const int* in_sizes, int n_in,
                              void* d_out, int out_size, void* d_ws,
                              size_t ws_size, hipStream_t stream) {
    const float* x         = (const float*)d_in[0];
    const int*   edge_idx  = (const int*)d_in[1];
    const float* edge_attr = (const float*)d_in[2];
    const float* bn_pre_w  = (const float*)d_in[3];
    const float* bn_pre_b  = (const float*)d_in[4];
    const float* edge_lin_w= (const float*)d_in[5];
    const float* edge_lin_b= (const float*)d_in[6];
    const float* lin1_w    = (const float*)d_in[7];
    const float* lin1_b    = (const float*)d_in[8];
    const float* bn_mid_w  = (const float*)d_in[9];
    const float* bn_mid_b  = (const float*)d_in[10];
    const float* lin2_w    = (const float*)d_in[11];
    const float* lin2_b    = (const float*)d_in[12];
    const float* epsArr    = (const float*)d_in[13];
    const float* cls1_w    = (const float*)d_in[14];
    const float* cls1_b    = (const float*)d_in[15];
    const float* cls_bn_w  = (const float*)d_in[16];
    const float* cls_bn_b  = (const float*)d_in[17];
    const float* cls2_w    = (const float*)d_in[18];
    const float* cls2_b    = (const float*)d_in[19];
    float* y = (float*)d_out;

    const int N = in_sizes[0] / D64;       // 100000
    const int E = in_sizes[2] / ED16;      // 1600000
    const int L = 4;
    const int* src = edge_idx;
    const int* dst = edge_idx + E;

    // Workspace layout (floats):
    //   h [N*64] | hi [N*64] | aggr [N*64] | z1 [N*64] | stats (512)
    //   classifier t [N*128] reuses (hi,aggr) contiguous region
    float* ws = (float*)d_ws;
    const size_t nd = (size_t)N * D64;
    float* h    = ws;
    float* hi   = ws + nd;
    float* aggr = ws + 2 * nd;
    float* z1   = ws + 3 * nd;
    float* gsum   = ws + 4 * nd;
    float* gsumsq = gsum + 128;
    float* scaleB = gsum + 256;
    float* shiftB = gsum + 384;
    float* tbuf = hi;   // [N*128], spans hi+aggr regions

    const int total = N * D64;
    const int ewBlocks = (total + 255) / 256;
    const int statsBlocks = 512;
    const int rowTiles = N / 16;                       // 6250
    const int gemmBlocks = (rowTiles + 7) / 8;
    const int edgeTiles = E / 16;                      // 100000
    const int edgeBlocks = (edgeTiles + 7) / 8;
    const int clsBlocks = (N + 7) / 8;

    for (int i = 0; i < L; i++) {
        const float* hin = (i == 0) ? x : h;
        // --- pre-BN over current h ---
        fill_zero_kernel<<<1, 256, 0, stream>>>(gsum, 256);
        bn_stats_kernel<D64><<<statsBlocks, 256, 0, stream>>>(hin, N, gsum, gsumsq);
        bn_finalize_kernel<<<1, D64, 0, stream>>>(gsum, gsumsq, N,
                                                  bn_pre_w + i * D64,
                                                  bn_pre_b + i * D64, scaleB, shiftB);
        // --- hi = act(bn(h)); aggr = (1+eps)*hi ---
        node_prep_kernel<<<ewBlocks, 256, 0, stream>>>(hin, scaleB, shiftB, epsArr,
                                                       i, hi, aggr, total);
        // --- edge message + scatter-add ---
        edge_kernel<<<edgeBlocks, 256, 0, stream>>>(
            edge_attr, src, dst, edge_lin_w + (size_t)i * D64 * ED16,
            edge_lin_b + i * D64, hi, aggr, edgeTiles);
        // --- z1 = aggr @ lin1^T + b1 ---
        gemm_wmma_kernel<D64><<<gemmBlocks, 256, 0, stream>>>(
            aggr, lin1_w + (size_t)i * D64 * D64, lin1_b + i * D64, nullptr,
            nullptr, nullptr, z1, rowTiles, 0);
        // --- mid-BN over z1 ---
        fill_zero_kernel<<<1, 256, 0, stream>>>(gsum, 256);
        bn_stats_kernel<D64><<<statsBlocks, 256, 0, stream>>>(z1, N, gsum, gsumsq);
        bn_finalize_kernel<<<1, D64, 0, stream>>>(gsum, gsumsq, N,
                                                  bn_mid_w + i * D64,
                                                  bn_mid_b + i * D64, scaleB, shiftB);
        // --- h = [h +] leaky(bn(z1)) @ lin2^T + b2 ---
        gemm_wmma_kernel<D64><<<gemmBlocks, 256, 0, stream>>>(
            z1, lin2_w + (size_t)i * D64 * D64, lin2_b + i * D64, scaleB, shiftB,
            (i == 0) ? nullptr : h, h, rowTiles, 1);
    }

    // --- classifier: t = h @ cls1^T + b1 ---
    gemm_wmma_kernel<M128><<<gemmBlocks, 256, 0, stream>>>(
        h, cls1_w, cls1_b, nullptr, nullptr, nullptr, tbuf, rowTiles, 0);
    fill_zero_kernel<<<1, 256, 0, stream>>>(gsum, 256);
    bn_stats_kernel<M128><<<statsBlocks, 256, 0, stream>>>(tbuf, N, gsum, gsumsq);
    bn_finalize_kernel<<<1, M128, 0, stream>>>(gsum, gsumsq, N, cls_bn_w, cls_bn_b,
                                               scaleB, shiftB);
    cls_final_kernel<<<clsBlocks, 256, 0, stream>>>(tbuf, scaleB, shiftB, cls2_w,
                                                    cls2_b, y, N);
}